// HyperTurbulentGenerator_88441966559955
// MI455X (gfx1250) — compile-verified
//
#include <hip/hip_runtime.h>

// HyperTurbulentGenerator — CDNA5 (gfx1250) implementation.
// out[b,s,:] = M^{s+1} (z0r + i z0i),  M = expm(A*dt + eps*I),
// A = alpha*(K - K^T) - beta*(R^T R),  K/R = basis contractions.
// One workgroup (256 threads = 8 wave32) per batch element; all matrices in LDS;
// every matmul runs on V_WMMA_F32_16X16X4_F32.

typedef __attribute__((ext_vector_type(2))) float v2f;
typedef __attribute__((ext_vector_type(8))) float v8f;

#define DD      128
#define NBASES  8
#define NSTEPS  128
#define LSTR    132                // padded LDS row stride (floats): bank-conflict-free frag loads
#define MATF    (DD * LSTR)        // 16896 floats per 128x128 LDS matrix
#define SCAL_K  6                  // expm scaling: B0 = (A*dt+eps I) * 2^-6
#define TAYLOR_N 10                // Taylor order (Horner)
#define EPS_DIAG 1e-6f

extern __shared__ float lds[];

// C[:, dst..dst+16*NCT) = outScale * (A or A^T) @ B[:, src..src+16*NCT)  [+ I if ADDI]
// A is a full 128x128 LDS matrix; B/C are (possibly the same) 128-row LDS matrices.
// 8 waves each own a 16-row strip; K consumed 4 at a time via wmma f32 16x16x4.
// All NCT B-fragments are staged into distinct registers before the WMMA burst so the
// scheduler can issue the ds_loads back-to-back with a single s_wait_dscnt.
template <int NCT, bool TRANSA, bool ADDI>
__device__ __forceinline__ void mm128(int pC, int pA, int pB,
                                      int srcColBase, int dstColBase, float outScale)
{
    const int wave = threadIdx.x >> 5;
    const int lane = threadIdx.x & 31;
    const int m = lane & 15;          // row-in-tile (A), col-in-tile (B/C)
    const int h = lane >> 4;          // K-pair half (A/B), row-half (C)
    const int rowBase = 16 * wave;

    v8f acc[NCT];
#pragma unroll
    for (int t = 0; t < NCT; ++t)
        acc[t] = v8f{0.f, 0.f, 0.f, 0.f, 0.f, 0.f, 0.f, 0.f};

    for (int k4 = 0; k4 < 32; ++k4) {
        const int c = 4 * k4 + 2 * h; // this lane's first K index of the pair
        v2f a;
        if (!TRANSA) {
            a = *(const v2f*)&lds[pA + (rowBase + m) * LSTR + c];
        } else {
            a.x = lds[pA + c * LSTR + rowBase + m];
            a.y = lds[pA + (c + 1) * LSTR + rowBase + m];
        }
        v2f bf[NCT];
#pragma unroll
        for (int t = 0; t < NCT; ++t) {
            const int n = srcColBase + 16 * t + m;
            bf[t].x = lds[pB + c * LSTR + n];
            bf[t].y = lds[pB + (c + 1) * LSTR + n];
        }
#pragma unroll
        for (int t = 0; t < NCT; ++t)
            acc[t] = __builtin_amdgcn_wmma_f32_16x16x4_f32(
                false, a, false, bf[t], (short)0, acc[t], false, false);
    }
#pragma unroll
    for (int t = 0; t < NCT; ++t) {
        const int cb = dstColBase + 16 * t + m;
#pragma unroll
        for (int r = 0; r < 8; ++r) { // C layout: VGPR r <-> row 8h+r
            const int row = rowBase + 8 * h + r;
            float v = outScale * acc[t][r];
            if (ADDI && row == cb) v += 1.0f;   // fused Horner "+ I"
            lds[pC + row * LSTR + cb] = v;
        }
    }
}

__global__ void __launch_bounds__(256, 1)
hyper_turbulent_evolve(const float* __restrict__ z0r, const float* __restrict__ z0i,
                       const float* __restrict__ time_steps,
                       const float* __restrict__ k_coeffs, const float* __restrict__ r_coeffs,
                       const float* __restrict__ alpha, const float* __restrict__ beta,
                       const float* __restrict__ K_bases, const float* __restrict__ R_bases,
                       float2* __restrict__ out)
{
    const int b   = blockIdx.x;
    const int tid = threadIdx.x;
    const int off0 = 0, off1 = MATF, off2 = 2 * MATF, off3 = 3 * MATF;
    float* sh = &lds[4 * MATF];       // small scalar staging area

    if (tid < NBASES) {
        sh[tid]     = k_coeffs[b * NBASES + tid];
        sh[8 + tid] = r_coeffs[b * NBASES + tid];
    }
    if (tid == 0) {
        sh[16] = alpha[b];
        sh[17] = beta[b];
        sh[18] = time_steps[0];       // uniform-grid branch: dt = ts[0]
    }
    __syncthreads();

    // ---- Phase 1: basis contractions. K_sum -> off1, R_sum -> off2 ----
    for (int idx = tid; idx < DD * DD; idx += 256) {
        const int row = idx >> 7, col = idx & 127;
        float ks = 0.f, rs = 0.f;
#pragma unroll
        for (int n = 0; n < NBASES; ++n) {
            ks += sh[n]     * K_bases[(n * DD + row) * DD + col];
            rs += sh[8 + n] * R_bases[(n * DD + row) * DD + col];
        }
        lds[off1 + row * LSTR + col] = ks;
        lds[off2 + row * LSTR + col] = rs;
    }
    __syncthreads();

    // ---- gram = -(R^T R) -> off3  (WMMA, transposed-A fragments) ----
    mm128<8, true, false>(off3, off2, off2, 0, 0, -1.0f);
    __syncthreads();

    // ---- B0 = (A*dt + eps I) * 2^-k -> off2 ;  A = alpha*(K-K^T) + beta*gram ----
    {
        const float al = sh[16], be = sh[17], dt = sh[18];
        const float sc   = dt * (1.0f / (float)(1 << SCAL_K));
        const float epsc = EPS_DIAG * (1.0f / (float)(1 << SCAL_K));
        for (int idx = tid; idx < DD * DD; idx += 256) {
            const int row = idx >> 7, col = idx & 127;
            const float Aij = al * (lds[off1 + row * LSTR + col] - lds[off1 + col * LSTR + row])
                            + be * lds[off3 + row * LSTR + col];
            lds[off2 + row * LSTR + col] = Aij * sc + ((row == col) ? epsc : 0.f);
        }
    }
    __syncthreads();

    // ---- expm(B0), Horner Taylor: Acc_{j-1} = I + (B0 @ Acc_j)/j ----
    // Seed with Acc_{N-1} = I + B0/N directly (skips the B0 @ I matmul).
    for (int idx = tid; idx < DD * DD; idx += 256) {
        const int row = idx >> 7, col = idx & 127;
        lds[off1 + row * LSTR + col] =
            lds[off2 + row * LSTR + col] * (1.0f / (float)TAYLOR_N)
            + ((row == col) ? 1.f : 0.f);
    }
    __syncthreads();

    int pAcc = off1, pT = off3;
    for (int j = TAYLOR_N - 1; j >= 1; --j) {          // 9 iterations (odd)
        mm128<8, false, true>(pT, off2, pAcc, 0, 0, 1.0f / (float)j);
        __syncthreads();
        const int t = pAcc; pAcc = pT; pT = t;
    }
    // pAcc = off3 after 9 swaps.
    // ---- squarings: M = exp(B0)^(2^k) ----  (SCAL_K = 6 even -> M back at off3? track below)
    for (int q = 0; q < SCAL_K; ++q) {
        mm128<8, false, false>(pT, pAcc, pAcc, 0, 0, 1.0f);
        __syncthreads();
        const int t = pAcc; pAcc = pT; pT = t;
    }
    const int pM = pAcc;              // ping-pong tracked; off1/off3 pair
    const int pTmp = pT;
    const int pWr = off0, pWi = off2; // result columns: col s = M^{s+1} z  (real / imag)

    // ---- seed columns 0..15 with sequential matvecs (VALU) ----
    {
        const int row = tid & 127, comp = tid >> 7;
        const int pW = comp ? pWi : pWr;
        const float* z = comp ? (z0i + (size_t)b * DD) : (z0r + (size_t)b * DD);
        float acc0 = 0.f;
        for (int k = 0; k < DD; ++k) acc0 += lds[pM + row * LSTR + k] * z[k];
        lds[pW + row * LSTR + 0] = acc0;
        __syncthreads();
        for (int s = 1; s < 16; ++s) {
            float a2 = 0.f;
            for (int k = 0; k < DD; ++k)
                a2 += lds[pM + row * LSTR + k] * lds[pW + k * LSTR + (s - 1)];
            lds[pW + row * LSTR + s] = a2;
            __syncthreads();
        }
    }

    // ---- P = M^16 via 4 squarings (M no longer needed afterwards) ----
    int pP = pM, pQ = pTmp;
    for (int q = 0; q < 4; ++q) {
        mm128<8, false, false>(pQ, pP, pP, 0, 0, 1.0f);
        __syncthreads();
        const int t = pP; pP = pQ; pQ = t;
    }

    // ---- column doubling: W[:, n..2n) = M^n @ W[:, 0..n) for n = 16, 32, 64 ----
    mm128<1, false, false>(pWr, pP, pWr, 0, 16, 1.0f);
    mm128<1, false, false>(pWi, pP, pWi, 0, 16, 1.0f);
    __syncthreads();
    mm128<8, false, false>(pQ, pP, pP, 0, 0, 1.0f);    // M^32
    __syncthreads();
    { const int t = pP; pP = pQ; pQ = t; }

    mm128<2, false, false>(pWr, pP, pWr, 0, 32, 1.0f);
    mm128<2, false, false>(pWi, pP, pWi, 0, 32, 1.0f);
    __syncthreads();
    mm128<8, false, false>(pQ, pP, pP, 0, 0, 1.0f);    // M^64
    __syncthreads();
    { const int t = pP; pP = pQ; pQ = t; }

    mm128<4, false, false>(pWr, pP, pWr, 0, 64, 1.0f);
    mm128<4, false, false>(pWi, pP, pWi, 0, 64, 1.0f);
    __syncthreads();

    // ---- emit complex64: out[b, s, o] = (Wr[o,s], Wi[o,s]) ----
    for (int idx = tid; idx < NSTEPS * DD; idx += 256) {
        const int s = idx >> 7, o = idx & 127;
        float2 v;
        v.x = lds[pWr + o * LSTR + s];
        v.y = lds[pWi + o * LSTR + s];
        out[((size_t)b * NSTEPS + s) * DD + o] = v;
    }
}

extern "C" void kernel_launch(void* const* d_in, const int* in_sizes, int n_in,
                              void* d_out, int out_size, void* d_ws, size_t ws_size,
                              hipStream_t stream)
{
    (void)n_in; (void)out_size; (void)d_ws; (void)ws_size;
    const float* z0r        = (const float*)d_in[0];
    const float* z0i        = (const float*)d_in[1];
    const float* time_steps = (const float*)d_in[2];
    const float* k_coeffs   = (const float*)d_in[3];
    const float* r_coeffs   = (const float*)d_in[4];
    const float* alpha      = (const float*)d_in[5];
    const float* beta       = (const float*)d_in[6];
    const float* K_bases    = (const float*)d_in[7];
    const float* R_bases    = (const float*)d_in[8];

    const int B = in_sizes[0] / DD;                       // 32
    const size_t shbytes = (size_t)(4 * MATF + 32) * sizeof(float); // ~270.5 KB of 320 KB WGP LDS

    hipFuncSetAttribute((const void*)hyper_turbulent_evolve,
                        hipFuncAttributeMaxDynamicSharedMemorySize, (int)shbytes);

    hyper_turbulent_evolve<<<dim3(B), dim3(256), shbytes, stream>>>(
        z0r, z0i, time_steps, k_coeffs, r_coeffs, alpha, beta,
        K_bases, R_bases, (float2*)d_out);
}